// Prediction_55508157334088
// MI455X (gfx1250) — compile-verified
//
#include <hip/hip_runtime.h>

typedef float v2f __attribute__((ext_vector_type(2)));
typedef float v4f __attribute__((ext_vector_type(4)));
typedef float v8f __attribute__((ext_vector_type(8)));

#define BAND_LEN 4096   // L (fixed by reference)
#define TPB      256    // 8 wave32 per workgroup

// One workgroup per 16-row stripe of one batch image.
// out[b,i,j] = s[b,i]*e[b,j] when 0 <= j-i <= 15 else 0.
__global__ __launch_bounds__(TPB) void banded_outer_kernel(
    const float* __restrict__ s, const float* __restrict__ e,
    float* __restrict__ out)
{
    const int L     = BAND_LEN;
    const int tiles = L / 16;                 // 256 row-blocks per batch
    const int blk   = blockIdx.x;             // 0 .. B*tiles-1
    const int b     = blk / tiles;
    const int t     = blk - b * tiles;
    const int r0    = t * 16;                 // first row of this stripe

    float* stripe = out + ((size_t)b * L + (size_t)r0) * (size_t)L; // 16 x L

    // --- 1) Non-temporal zero fill of the stripe, skipping the band window.
    // Band columns [c0, c1) are produced by wave 0 below (disjoint addresses,
    // so no barrier is required).
    const int c0  = r0;
    const int c1  = (r0 + 32 <= L) ? (r0 + 32) : L;
    const int q0  = c0 >> 2;                  // band window in float4 units
    const int q1  = c1 >> 2;
    const int qpr = L >> 2;                   // float4 slots per row (1024)

    const v4f zero4 = {0.f, 0.f, 0.f, 0.f};
    for (int idx = threadIdx.x; idx < 16 * (BAND_LEN / 4); idx += TPB) {
        const int q = idx & (qpr - 1);        // float4 column slot
        if (q >= q0 && q < q1) continue;      // band tiles: written by wave 0
        __builtin_nontemporal_store(zero4, (v4f*)stripe + idx);
    }

    // --- 2) Wave 0: band tiles via V_WMMA_F32_16X16X4_F32 (rank-1 product).
    if (threadIdx.x < 32) {
        const int  lane   = threadIdx.x;      // wave32: EXEC all-ones here
        const int  lo     = lane & 15;
        const bool lohalf = lane < 16;

        // A: 16x4 f32. ISA layout: lanes 0-15 -> {VGPR0=K0, VGPR1=K1},
        // lanes 16-31 -> {K2, K3}. Put s in the K=0 column only.
        v2f A;
        A.x = lohalf ? s[(size_t)b * L + r0 + lo] : 0.f;
        A.y = 0.f;

        // B: 4x16 f32, row-striped: VGPR0 lanes 0-15 = K0 row (e values).
        v2f B0;
        B0.x = lohalf ? e[(size_t)b * L + r0 + lo] : 0.f;
        B0.y = 0.f;

        const v8f czero = {0.f,0.f,0.f,0.f,0.f,0.f,0.f,0.f};

        // D[M,N] = sum_K A[M,K]*B[K,N] = s[r0+M] * e[r0+N]
        v8f d0 = __builtin_amdgcn_wmma_f32_16x16x4_f32(
            false, A, false, B0, (short)0, czero, false, false);

        // C/D layout: VGPR v, lanes 0-15 -> M=v ; lanes 16-31 -> M=v+8 ; N=lane&15
        const int N  = lo;
        const int Mb = (lane >> 4) << 3;

        // Diagonal tile: keep j>=i  <=>  N >= M  (j-i<=15 automatic in-tile).
        float* dst0 = stripe + (size_t)Mb * L + (size_t)(r0 + N);
        #pragma unroll
        for (int v = 0; v < 8; ++v) {
            const int   M   = Mb + v;
            const float val = (N >= M) ? d0[v] : 0.f;
            dst0[(size_t)v * L] = val;
        }

        // Off-diagonal tile (cols r0+16 .. r0+31): j-i = 16+N-M <= 15 <=> N < M.
        if (r0 + 16 < L) {
            v2f B1;
            B1.x = lohalf ? e[(size_t)b * L + r0 + 16 + lo] : 0.f;
            B1.y = 0.f;

            v8f d1 = __builtin_amdgcn_wmma_f32_16x16x4_f32(
                false, A, false, B1, (short)0, czero, false, false);

            float* dst1 = stripe + (size_t)Mb * L + (size_t)(r0 + 16 + N);
            #pragma unroll
            for (int v = 0; v < 8; ++v) {
                const int   M   = Mb + v;
                const float val = (N < M) ? d1[v] : 0.f;
                dst1[(size_t)v * L] = val;
            }
        }
    }
}

extern "C" void kernel_launch(void* const* d_in, const int* in_sizes, int n_in,
                              void* d_out, int out_size, void* d_ws, size_t ws_size,
                              hipStream_t stream) {
    (void)n_in; (void)out_size; (void)d_ws; (void)ws_size;
    const float* s = (const float*)d_in[0];
    const float* e = (const float*)d_in[1];
    float* out = (float*)d_out;

    const int L = BAND_LEN;
    const int B = in_sizes[0] / L;            // 8 for the reference shapes
    const int nblocks = B * (L / 16);         // one block per 16-row stripe

    banded_outer_kernel<<<nblocks, TPB, 0, stream>>>(s, e, out);
}